// GraphVAE_28810640621902
// MI455X (gfx1250) — compile-verified
//
#include <hip/hip_runtime.h>

// ---------------- problem sizes (from reference) ----------------
#define N_NODES    50000
#define N_EDGES    640000
#define NODE_DIM   16
#define HIDDEN     128
#define LATENT     64
#define MAX_NODES  100
#define NUM_GRAPHS 256

#define WPB 8   // waves per block in the GEMM (8 x wave32 = 256 threads)

typedef float v2f __attribute__((ext_vector_type(2)));
typedef float v8f __attribute__((ext_vector_type(8)));

// ---------------- elementwise helpers ----------------
__global__ void gv_zero_f32(float* __restrict__ p, int n) {
    for (int i = blockIdx.x * blockDim.x + threadIdx.x; i < n;
         i += gridDim.x * blockDim.x)
        p[i] = 0.0f;
}

__global__ void gv_copy_f32(float* __restrict__ dst, const float* __restrict__ src, int n) {
    for (int i = blockIdx.x * blockDim.x + threadIdx.x; i < n;
         i += gridDim.x * blockDim.x)
        dst[i] = src[i];
}

// agg[dst[e]][d] += feat[src[e]][d]   (agg pre-initialized with feat itself => GIN eps=0)
__global__ void gv_scatter_add(const float* __restrict__ feat,
                               const int* __restrict__ src,
                               const int* __restrict__ dst,
                               float* __restrict__ agg, int dim) {
    int total = N_EDGES * dim;   // <= 81.92M, fits int32
    for (int i = blockIdx.x * blockDim.x + threadIdx.x; i < total;
         i += gridDim.x * blockDim.x) {
        int e = i / dim;
        int d = i - e * dim;
        atomicAdd(&agg[(long)dst[e] * dim + d], feat[(long)src[e] * dim + d]);
    }
}

// ---------------- WMMA f32 GEMM: D = act(A(MxK) * B(KxN) + bias) ----------------
// One 16x16 output tile per wave32, 8 waves/block sharing one 16-column B
// panel staged in LDS. K templated -> fully unrolled chain of
// V_WMMA_F32_16X16X4_F32 (K/4 per tile). Requires M%16==0, N%16==0, K%4==0
// (true for every GEMM in this network). Out-of-range waves still run the
// staging loop + barrier; only compute/store is guarded, so EXEC is all-ones
// inside every WMMA wave as the ISA requires.
template <int ACT, int K>   // ACT: 0 = identity, 1 = relu
__global__ void gv_wmma_gemm(const float* __restrict__ A,
                             const float* __restrict__ B,
                             const float* __restrict__ bias,
                             float* __restrict__ D,
                             int M, int N) {
    __shared__ float Bs[K * 16];             // K x 16 B panel (<= 8 KB)

    const int n0     = blockIdx.x << 4;                       // N-tile origin
    const int mtile  = blockIdx.y * WPB + threadIdx.y;        // wave-uniform
    const int mtiles = M >> 4;
    const int m0     = mtile << 4;

    const int lane = threadIdx.x;            // 0..31
    const int l15  = lane & 15;
    const int kh   = (lane >> 4) << 1;       // 0 (lanes 0-15) or 2 (lanes 16-31)

    const float* arow = A + (long)(m0 + l15) * K + kh;
    const bool   live = (mtile < mtiles);
    if (live) __builtin_prefetch(arow);      // global_prefetch_b8: A stream in flight

    // cooperative global -> LDS stage of the B panel (all 256 threads)
    const int tid = threadIdx.y * 32 + lane;
#pragma unroll
    for (int i = tid; i < K * 16; i += WPB * 32)
        Bs[i] = B[(long)(i >> 4) * N + n0 + (i & 15)];
    __syncthreads();

    if (!live) return;

    const int ncol = n0 + l15;
    v8f c = {};
#pragma unroll
    for (int k = 0; k < K; k += 4) {
        // A 16x4 fragment: VGPR0 = K{k | k+2}, VGPR1 = K{k+1 | k+3}
        float2 t = *reinterpret_cast<const float2*>(arow + k);
        v2f a; a.x = t.x; a.y = t.y;
        // B 4x16 fragment from LDS: VGPR0 = row (k+kh), VGPR1 = row (k+kh+1)
        v2f b;
        b.x = Bs[(k + kh) * 16 + l15];
        b.y = Bs[(k + kh + 1) * 16 + l15];
        c = __builtin_amdgcn_wmma_f32_16x16x4_f32(
                /*neg_a=*/false, a, /*neg_b=*/false, b,
                /*c_mod=*/(short)0, c, /*reuse_a=*/false, /*reuse_b=*/false);
    }

    const float bv    = bias[ncol];
    const int   rbase = m0 + ((lane >> 4) << 3);   // rows m0..m0+7 | m0+8..m0+15
#pragma unroll
    for (int i = 0; i < 8; ++i) {
        float v = c[i] + bv;
        if (ACT == 1) v = fmaxf(v, 0.0f);
        D[(long)(rbase + i) * N + ncol] = v;
    }
}

// ---------------- pooling ----------------
__global__ void gv_pool_accum(const float* __restrict__ h,
                              const int* __restrict__ batch,
                              float* __restrict__ summed,
                              float* __restrict__ counts) {
    const int total = N_NODES * HIDDEN;
    for (int i = blockIdx.x * blockDim.x + threadIdx.x; i < total;
         i += gridDim.x * blockDim.x) {
        int n = i >> 7;          // /HIDDEN
        int d = i & (HIDDEN - 1);
        int g = batch[n];
        atomicAdd(&summed[g * HIDDEN + d], h[i]);
        if (d == 0) atomicAdd(&counts[g], 1.0f);
    }
}

__global__ void gv_pool_div(const float* __restrict__ summed,
                            const float* __restrict__ counts,
                            float* __restrict__ pooled) {
    const int total = NUM_GRAPHS * HIDDEN;
    for (int i = blockIdx.x * blockDim.x + threadIdx.x; i < total;
         i += gridDim.x * blockDim.x) {
        int g = i >> 7;
        pooled[i] = summed[i] / fmaxf(counts[g], 1.0f);
    }
}

// ---------------- latent: clip(logvar), z = mu + eps*exp(0.5*logvar) ----------------
__global__ void gv_latent(const float* __restrict__ mu,
                          const float* __restrict__ lv_raw,
                          const float* __restrict__ eps,
                          float* __restrict__ logvar_out,
                          float* __restrict__ z) {
    const int total = NUM_GRAPHS * LATENT;
    for (int i = blockIdx.x * blockDim.x + threadIdx.x; i < total;
         i += gridDim.x * blockDim.x) {
        float lv = fminf(fmaxf(lv_raw[i], -5.0f), 5.0f);
        logvar_out[i] = lv;
        z[i] = mu[i] + eps[i] * expf(0.5f * lv);
    }
}

// ---------------- softmax over last dim (16) of (256,100,16) ----------------
__global__ void gv_softmax16(const float* __restrict__ logits,
                             float* __restrict__ out) {
    const int rows = NUM_GRAPHS * MAX_NODES;   // 25600
    for (int r = blockIdx.x * blockDim.x + threadIdx.x; r < rows;
         r += gridDim.x * blockDim.x) {
        const float* p = logits + (long)r * NODE_DIM;
        float mx = p[0];
#pragma unroll
        for (int d = 1; d < NODE_DIM; ++d) mx = fmaxf(mx, p[d]);
        float e[NODE_DIM];
        float s = 0.0f;
#pragma unroll
        for (int d = 0; d < NODE_DIM; ++d) { e[d] = expf(p[d] - mx); s += e[d]; }
        float inv = 1.0f / s;
#pragma unroll
        for (int d = 0; d < NODE_DIM; ++d) out[(long)r * NODE_DIM + d] = e[d] * inv;
    }
}

// ---------------- adjacency epilogue: symmetrize, diag=-10, clip ----------------
__global__ void gv_adj_post(const float* __restrict__ raw, float* __restrict__ out) {
    const int total = NUM_GRAPHS * MAX_NODES * MAX_NODES;
    for (int i = blockIdx.x * blockDim.x + threadIdx.x; i < total;
         i += gridDim.x * blockDim.x) {
        int g   = i / (MAX_NODES * MAX_NODES);
        int rem = i - g * (MAX_NODES * MAX_NODES);
        int r   = rem / MAX_NODES;
        int cc  = rem - r * MAX_NODES;
        const float* base = raw + (long)g * MAX_NODES * MAX_NODES;
        float v = 0.5f * (base[r * MAX_NODES + cc] + base[cc * MAX_NODES + r]);
        if (r == cc) v = -10.0f;
        v = fminf(fmaxf(v, -10.0f), 10.0f);
        out[i] = v;
    }
}

// ---------------- launch ----------------
static inline int gv_cdiv(int a, int b) { return (a + b - 1) / b; }

extern "C" void kernel_launch(void* const* d_in, const int* in_sizes, int n_in,
                              void* d_out, int out_size, void* d_ws, size_t ws_size,
                              hipStream_t stream) {
    (void)in_sizes; (void)n_in; (void)out_size; (void)ws_size;

    const float* x    = (const float*)d_in[0];
    const int*   edge = (const int*)d_in[1];
    const int*   batch= (const int*)d_in[2];
    const float* eps  = (const float*)d_in[3];
    const float* w1a  = (const float*)d_in[4];  const float* b1a = (const float*)d_in[5];
    const float* w1b  = (const float*)d_in[6];  const float* b1b = (const float*)d_in[7];
    const float* w2a  = (const float*)d_in[8];  const float* b2a = (const float*)d_in[9];
    const float* w2b  = (const float*)d_in[10]; const float* b2b = (const float*)d_in[11];
    const float* wmu  = (const float*)d_in[12]; const float* bmu = (const float*)d_in[13];
    const float* wlv  = (const float*)d_in[14]; const float* blv = (const float*)d_in[15];
    const float* wd1  = (const float*)d_in[16]; const float* bd1 = (const float*)d_in[17];
    const float* wd2  = (const float*)d_in[18]; const float* bd2 = (const float*)d_in[19];
    const float* wn   = (const float*)d_in[20]; const float* bn  = (const float*)d_in[21];
    const float* we   = (const float*)d_in[22]; const float* be  = (const float*)d_in[23];

    const int* srcI = edge;              // edge_index[0]
    const int* dstI = edge + N_EDGES;    // edge_index[1]

    // ---- workspace layout (floats); peak 20,000,000 f32 = 80 MB ----
    float* ws   = (float*)d_ws;
    float* agg1 = ws;                       //   800,000 (50000x16)
    float* bufA = ws +   800000;            // 6,400,000 (50000x128)
    float* bufB = ws +  7200000;            // 6,400,000
    float* bufC = ws + 13600000;            // 6,400,000
    // decoder scratch aliases bufA (free after h2) / bufC (free after h2a)
    float* summed = bufA;                   //  32768
    float* counts = bufA + 32768;           //    256
    float* pooled = bufA + 33024;           //  32768
    float* lv_raw = bufA + 65792;           //  16384
    float* zbuf   = bufA + 82176;           //  16384
    float* hd1    = bufA + 98560;           //  32768
    float* hd     = bufA + 131328;          //  32768
    float* nlog   = bufA + 164096;          // 409600
    float* alog   = bufC;                   // 2,560,000

    // ---- output layout (return order: adj, node_features, mu, logvar) ----
    float* out     = (float*)d_out;
    float* out_adj = out;                               // 2,560,000
    float* out_nf  = out + 2560000;                     //   409,600
    float* out_mu  = out + 2560000 + 409600;            //    16,384
    float* out_lv  = out_mu + 16384;                    //    16,384

    const int T = 256;
    dim3 gblk(32, WPB);   // 8 waves per block, one 16x16 tile per wave
    auto ggrid = [](int M, int N) {
        return dim3((unsigned)(N / 16), (unsigned)gv_cdiv(M / 16, WPB));
    };

    // ===== encoder layer 1 =====
    gv_copy_f32<<<gv_cdiv(N_NODES * NODE_DIM, T), T, 0, stream>>>(agg1, x, N_NODES * NODE_DIM);
    gv_scatter_add<<<16384, T, 0, stream>>>(x, srcI, dstI, agg1, NODE_DIM);
    gv_wmma_gemm<1, NODE_DIM><<<ggrid(N_NODES, HIDDEN), gblk, 0, stream>>>(agg1, w1a, b1a, bufA, N_NODES, HIDDEN);
    gv_wmma_gemm<1, HIDDEN><<<ggrid(N_NODES, HIDDEN), gblk, 0, stream>>>(bufA, w1b, b1b, bufB, N_NODES, HIDDEN);

    // ===== encoder layer 2 =====
    gv_copy_f32<<<gv_cdiv(N_NODES * HIDDEN, T), T, 0, stream>>>(bufC, bufB, N_NODES * HIDDEN);
    gv_scatter_add<<<65536, T, 0, stream>>>(bufB, srcI, dstI, bufC, HIDDEN);
    gv_wmma_gemm<1, HIDDEN><<<ggrid(N_NODES, HIDDEN), gblk, 0, stream>>>(bufC, w2a, b2a, bufA, N_NODES, HIDDEN);
    gv_wmma_gemm<1, HIDDEN><<<ggrid(N_NODES, HIDDEN), gblk, 0, stream>>>(bufA, w2b, b2b, bufB, N_NODES, HIDDEN);

    // ===== mean pool per graph =====
    gv_zero_f32<<<gv_cdiv(33024, T), T, 0, stream>>>(summed, 33024);  // summed + counts
    gv_pool_accum<<<32768, T, 0, stream>>>(bufB, batch, summed, counts);
    gv_pool_div<<<gv_cdiv(NUM_GRAPHS * HIDDEN, T), T, 0, stream>>>(summed, counts, pooled);

    // ===== latent heads + reparameterize =====
    gv_wmma_gemm<0, HIDDEN><<<ggrid(NUM_GRAPHS, LATENT), gblk, 0, stream>>>(pooled, wmu, bmu, out_mu, NUM_GRAPHS, LATENT);
    gv_wmma_gemm<0, HIDDEN><<<ggrid(NUM_GRAPHS, LATENT), gblk, 0, stream>>>(pooled, wlv, blv, lv_raw, NUM_GRAPHS, LATENT);
    gv_latent<<<gv_cdiv(NUM_GRAPHS * LATENT, T), T, 0, stream>>>(out_mu, lv_raw, eps, out_lv, zbuf);

    // ===== decoder =====
    gv_wmma_gemm<1, LATENT><<<ggrid(NUM_GRAPHS, HIDDEN), gblk, 0, stream>>>(zbuf, wd1, bd1, hd1, NUM_GRAPHS, HIDDEN);
    gv_wmma_gemm<0, HIDDEN><<<ggrid(NUM_GRAPHS, HIDDEN), gblk, 0, stream>>>(hd1, wd2, bd2, hd, NUM_GRAPHS, HIDDEN);

    gv_wmma_gemm<0, HIDDEN><<<ggrid(NUM_GRAPHS, MAX_NODES * NODE_DIM), gblk, 0, stream>>>(hd, wn, bn, nlog, NUM_GRAPHS, MAX_NODES * NODE_DIM);
    gv_softmax16<<<gv_cdiv(NUM_GRAPHS * MAX_NODES, T), T, 0, stream>>>(nlog, out_nf);

    gv_wmma_gemm<0, HIDDEN><<<ggrid(NUM_GRAPHS, MAX_NODES * MAX_NODES), gblk, 0, stream>>>(hd, we, be, alog, NUM_GRAPHS, MAX_NODES * MAX_NODES);
    gv_adj_post<<<gv_cdiv(NUM_GRAPHS * MAX_NODES * MAX_NODES, T), T, 0, stream>>>(alog, out_adj);
}